// IzhikevichNeuron_40518721470626
// MI455X (gfx1250) — compile-verified
//
#include <hip/hip_runtime.h>
#include <stdint.h>

// Izhikevich SNN scan on gfx1250 (MI455X).
// 16384 independent (batch, neuron) chains, 2000 serial steps.
// Memory-bound: ~512 MB traffic -> ~22 us floor @ 23.3 TB/s; with only
// 512 wave32s (1/SIMD) the per-step issue count is the runtime, budget
// ~25-30 slots/step. State (v,u) lives in registers; inputs stream through
// a 32-deep LDS ring filled by GLOBAL_LOAD_ASYNC_TO_LDS_B32 (SADDR form,
// in-order ASYNCcnt completion); outputs leave via SADDR-form non-temporal
// stores. Manual 4-step unroll: one s_wait_asynccnt + one s_wait_dscnt per
// 4 steps, immediate-folded lookahead reissues.

#define NSTEPS 2000
#define NNEUR  512
#define NBATCH 32
#define NTOT   (NBATCH * NSTEPS * NNEUR)   // 32,768,000 elements per output
#define DEPTH  32                          // ring stages (pow2, <= 63 ASYNCcnt)
#define BLK    256
#define UNROLL 4

__global__ __launch_bounds__(BLK) void izhikevich_scan_kernel(
    const float* __restrict__ in, float* __restrict__ out)
{
    __shared__ float ring[DEPTH * BLK];    // 32 KB ring: DEPTH stages x 256 lanes

    const int tid = threadIdx.x;
    const int blk = blockIdx.x;            // 0..63
    const int bb  = blk >> 1;              // batch (uniform per block)
    const int n0  = (blk & 1) << 8;        // neuron base 0 or 256 (uniform)

    // Wave-uniform base pointers -> SGPR pairs -> SADDR-form vmem.
    const float* gin_blk  = in  + (size_t)bb * (NSTEPS * NNEUR) + n0;
    float*       outS_blk = out + (size_t)bb * (NSTEPS * NNEUR) + n0;
    float*       outV_blk = outS_blk + (size_t)NTOT;
    float*       outU_blk = outS_blk + (size_t)2 * NTOT;

    // Hardware LDS byte offset of this lane's ring slot (low 32 bits of the
    // generic pointer are the LDS offset per the shared-aperture mapping).
    const unsigned ldsBase = (unsigned)(uintptr_t)(&ring[tid]);

    // ---- Prologue: launch DEPTH async prefetches (ASYNCcnt -> DEPTH) ----
    #pragma unroll
    for (int p = 0; p < DEPTH; ++p) {
        unsigned lws  = ldsBase + (unsigned)(p * BLK * 4);
        unsigned goff = (unsigned)(p * NNEUR * 4 + tid * 4);   // byte offset
        asm volatile("global_load_async_to_lds_b32 %0, %1, %2"
                     :: "v"(lws), "v"(goff), "s"(gin_blk) : "memory");
    }

    float v = -65.0f;                  // C
    float u = -13.0f;                  // B*C
    unsigned boff = (unsigned)(tid * 4);   // byte offset of step-t input element

    for (int t = 0; t < NSTEPS; t += UNROLL) {
        // Before this body: issued = t + DEPTH. Async loads complete in order,
        // so <= DEPTH-UNROLL outstanding <=> stages t..t+3 have all landed.
        asm volatile("s_wait_asynccnt 28" ::: "memory");

        const int stage = t & (DEPTH - 1);            // multiple of 4
        const float* rb = &ring[(stage << 8) + tid];
        float iv[UNROLL];
        #pragma unroll
        for (int j = 0; j < UNROLL; ++j) iv[j] = rb[j * BLK];  // ds offsets fold

        float sA[UNROLL], vA[UNROLL], uA[UNROLL];
        #pragma unroll
        for (int j = 0; j < UNROLL; ++j) {
            // poly = 0.04 v^2 + 5 v + 140 (pre-update v)
            float poly = fmaf(0.04f * v, v, fmaf(5.0f, v, 140.0f));
            // v' = v + 0.5*(poly - u + I); u' = u + 0.01*(0.2 v - u)
            float vn = fmaf(0.5f, poly, fmaf(-0.5f, u, fmaf(0.5f, iv[j], v)));
            float un = fmaf(0.002f, v, 0.99f * u);
            const bool fired = (vn >= 30.0f);
            const float s = fired ? 1.0f : 0.0f;
            v = fired ? -65.0f : vn;                  // v*(1-s) + C*s
            u = fmaf(8.0f, s, un);                    // u + D*s
            sA[j] = s; vA[j] = v; uA[j] = u;
        }

        const unsigned widx = (unsigned)t * NNEUR + (unsigned)tid;
        #pragma unroll
        for (int j = 0; j < UNROLL; ++j) {
            __builtin_nontemporal_store(sA[j], outS_blk + widx + j * NNEUR);
            __builtin_nontemporal_store(vA[j], outV_blk + widx + j * NNEUR);
            __builtin_nontemporal_store(uA[j], outU_blk + widx + j * NNEUR);
        }

        // Recycle stages t..t+3 for steps t+32..t+35: same lane offsets,
        // lookahead folded into 24-bit immediates (65536 + j*2048 bytes).
        if (t < NSTEPS - DEPTH) {                     // uniform scalar branch
            asm volatile("s_wait_dscnt 0x0" ::: "memory");  // ds_loads retired
            unsigned lws = ldsBase + (unsigned)(stage << 10);
            asm volatile("global_load_async_to_lds_b32 %0, %1, %2 offset:65536"
                         :: "v"(lws), "v"(boff), "s"(gin_blk) : "memory");
            unsigned l1 = lws + 1024u;
            asm volatile("global_load_async_to_lds_b32 %0, %1, %2 offset:67584"
                         :: "v"(l1), "v"(boff), "s"(gin_blk) : "memory");
            unsigned l2 = lws + 2048u;
            asm volatile("global_load_async_to_lds_b32 %0, %1, %2 offset:69632"
                         :: "v"(l2), "v"(boff), "s"(gin_blk) : "memory");
            unsigned l3 = lws + 3072u;
            asm volatile("global_load_async_to_lds_b32 %0, %1, %2 offset:71680"
                         :: "v"(l3), "v"(boff), "s"(gin_blk) : "memory");
        }
        boff += (unsigned)(UNROLL * NNEUR * 4);
    }
}

extern "C" void kernel_launch(void* const* d_in, const int* in_sizes, int n_in,
                              void* d_out, int out_size, void* d_ws, size_t ws_size,
                              hipStream_t stream) {
    (void)in_sizes; (void)n_in; (void)d_ws; (void)ws_size; (void)out_size;
    const float* in = (const float*)d_in[0];
    float* out = (float*)d_out;
    // 16384 chains / 256 threads = 64 blocks (8 wave32 per block).
    izhikevich_scan_kernel<<<dim3((NBATCH * NNEUR) / BLK), dim3(BLK), 0, stream>>>(in, out);
}